// HGT_90366111908558
// MI455X (gfx1250) — compile-verified
//
#include <hip/hip_runtime.h>
#include <math.h>

typedef float v2f __attribute__((ext_vector_type(2)));
typedef float v8f __attribute__((ext_vector_type(8)));

// ---------------------------------------------------------------------------
// Projection: P = E @ Wq  (rows x 64) @ (64 x 64), fp32 WMMA 16x16x4.
// One wave computes a 16-row x 64-col tile (4 accumulators of 16x16).
// A-frag (16x4 f32): lane<16 holds A[r][0..1], lane>=16 holds A[r][2..3].
// B-frag (4x16 f32): lane<16 holds B[0..1][r], lane>=16 holds B[2..3][r].
// C/D: VGPR j -> M=j (lanes 0-15) / M=j+8 (lanes 16-31), N=lane%16.
//
// W_Q is staged into LDS pre-swizzled into B-fragment order so that each
// fragment is one contiguous ds_load_b64 (no strided 2addr loads / v_movs):
//   frag index p = ((k0/4)*8 + kh*4 + nt)*16 + r   -> float2 {Wq[k][c], Wq[k+1][c]}
//   with k = k0 + 2*kh, c = nt*16 + r.
// ---------------------------------------------------------------------------
__global__ __launch_bounds__(256)
void proj_wmma_kernel(const float* __restrict__ E, const float* __restrict__ Wq,
                      float* __restrict__ P, int n_rows) {
  __shared__ float2 lB[2048];                 // 16 KB, B-fragment order
  for (int p = threadIdx.x; p < 2048; p += 256) {
    const int r  = p & 15;
    const int nt = (p >> 4) & 3;
    const int kh = (p >> 6) & 1;
    const int kq = p >> 7;                    // k0/4
    const int k  = kq * 4 + kh * 2;
    const int c  = nt * 16 + r;
    float2 f;
    f.x = Wq[(size_t)k * 64 + c];
    f.y = Wq[(size_t)(k + 1) * 64 + c];
    lB[p] = f;
  }
  __syncthreads();

  const int wave = threadIdx.x >> 5;
  const int lane = threadIdx.x & 31;
  const int r    = lane & 15;
  const int kh   = lane >> 4;                 // which K-half this lane holds
  const int rowBase = (blockIdx.x * 8 + wave) * 16;
  if (rowBase >= n_rows) return;              // wave-uniform: EXEC stays all-1s

  v8f acc0 = {0,0,0,0,0,0,0,0};
  v8f acc1 = acc0, acc2 = acc0, acc3 = acc0;

  const int arow = min(rowBase + r, n_rows - 1);   // clamp keeps EXEC full
  const float* aptr = E + (size_t)arow * 64 + kh * 2;
  const float2* bptr = lB + kh * 64 + r;      // + (k0/4)*128 + nt*16 per frag

  for (int k0 = 0; k0 < 64; k0 += 4) {
    v2f a;
    a.x = aptr[k0];
    a.y = aptr[k0 + 1];
    const float2* bq = bptr + (k0 >> 2) * 128;
    float2 f0 = bq[0];
    float2 f1 = bq[16];
    float2 f2 = bq[32];
    float2 f3 = bq[48];
    v2f b;
    b.x = f0.x; b.y = f0.y;
    acc0 = __builtin_amdgcn_wmma_f32_16x16x4_f32(false, a, false, b, (short)0, acc0, false, false);
    b.x = f1.x; b.y = f1.y;
    acc1 = __builtin_amdgcn_wmma_f32_16x16x4_f32(false, a, false, b, (short)0, acc1, false, false);
    b.x = f2.x; b.y = f2.y;
    acc2 = __builtin_amdgcn_wmma_f32_16x16x4_f32(false, a, false, b, (short)0, acc2, false, false);
    b.x = f3.x; b.y = f3.y;
    acc3 = __builtin_amdgcn_wmma_f32_16x16x4_f32(false, a, false, b, (short)0, acc3, false, false);
  }

  #pragma unroll
  for (int j = 0; j < 8; ++j) {
    const int row = rowBase + j + kh * 8;
    if (row < n_rows) {
      float* o = P + (size_t)row * 64 + r;
      o[0]  = acc0[j];
      o[16] = acc1[j];
      o[32] = acc2[j];
      o[48] = acc3[j];
    }
  }
}

// ---------------------------------------------------------------------------
// Per-edge attention logits + segment max (one edge per wave32).
// lane owns channels [2*lane, 2*lane+1]; head = lane>>4. Half-wave reduce.
// ---------------------------------------------------------------------------
__device__ __forceinline__ int enc_ordered(float f) {
  int i = __float_as_int(f);
  return i >= 0 ? i : (i ^ 0x7fffffff);
}
__device__ __forceinline__ float dec_ordered(int i) {
  return i >= 0 ? __int_as_float(i) : __int_as_float(i ^ 0x7fffffff);
}

__global__ __launch_bounds__(256)
void score_kernel(const float* __restrict__ P, const float* __restrict__ relE,
                  const int* __restrict__ head, const int* __restrict__ tail,
                  const int* __restrict__ etype,
                  float* __restrict__ scorebuf, int* __restrict__ maxbuf, int nedge) {
  const int e = blockIdx.x * 8 + (threadIdx.x >> 5);
  if (e >= nedge) return;
  const int lane = threadIdx.x & 31;
  const int h = head[e], t = tail[e], rt = etype[e] - 1;
  const float2 q = ((const float2*)(P    + (size_t)h  * 64))[lane];
  const float2 k = ((const float2*)(P    + (size_t)t  * 64))[lane];
  const float2 r = ((const float2*)(relE + (size_t)rt * 64))[lane];
  float p = q.x * k.x * r.x + q.y * k.y * r.y;
  p += __shfl_xor(p, 1, 32);
  p += __shfl_xor(p, 2, 32);
  p += __shfl_xor(p, 4, 32);
  p += __shfl_xor(p, 8, 32);      // reduced within each 16-lane half (one head)
  p *= 0.17677669529663687f;      // 1/sqrt(32)
  if ((lane & 15) == 0) {
    const int hh = lane >> 4;
    scorebuf[(size_t)e * 2 + hh] = p;
    atomicMax(maxbuf + (size_t)h * 2 + hh, enc_ordered(p));
  }
}

// ---------------------------------------------------------------------------
// ex = exp(score - segmax); denom[seg] += ex. One thread per (edge, head).
// ---------------------------------------------------------------------------
__global__ __launch_bounds__(256)
void exp_kernel(float* __restrict__ scorebuf, const int* __restrict__ head,
                const int* __restrict__ maxbuf, float* __restrict__ denom, int nedge) {
  const int i = blockIdx.x * 256 + threadIdx.x;
  if (i >= nedge * 2) return;
  const int e = i >> 1, hh = i & 1;
  const int h = head[e];
  const float mf = dec_ordered(maxbuf[(size_t)h * 2 + hh]);
  const float ex = __expf(scorebuf[i] - mf);
  scorebuf[i] = ex;
  atomicAdd(denom + (size_t)h * 2 + hh, ex);
}

// ---------------------------------------------------------------------------
// Weighted message scatter-add: aggE[head] += (E[tail]*rel) * attn. Edge/wave.
// ---------------------------------------------------------------------------
__global__ __launch_bounds__(256)
void agg_kernel(const float* __restrict__ Ein, const float* __restrict__ relE,
                const int* __restrict__ head, const int* __restrict__ tail,
                const int* __restrict__ etype, const float* __restrict__ scorebuf,
                const float* __restrict__ denom, float* __restrict__ aggE, int nedge) {
  const int e = blockIdx.x * 8 + (threadIdx.x >> 5);
  if (e >= nedge) return;
  const int lane = threadIdx.x & 31;
  const int hh = lane >> 4;
  const int h = head[e], t = tail[e], rt = etype[e] - 1;
  const float attn = scorebuf[(size_t)e * 2 + hh] / denom[(size_t)h * 2 + hh];
  const float2 v = ((const float2*)(Ein  + (size_t)t  * 64))[lane];
  const float2 r = ((const float2*)(relE + (size_t)rt * 64))[lane];
  float* o = aggE + (size_t)h * 64 + lane * 2;
  atomicAdd(o,     v.x * r.x * attn);
  atomicAdd(o + 1, v.y * r.y * attn);
}

// ---------------------------------------------------------------------------
// User aggregation: aggU[user] += w * E[item]. One UI-edge per wave.
// ---------------------------------------------------------------------------
__global__ __launch_bounds__(256)
void user_agg_kernel(const float* __restrict__ Ein, const int* __restrict__ iu,
                     const int* __restrict__ ii, const float* __restrict__ w,
                     float* __restrict__ aggU, int nui) {
  const int e = blockIdx.x * 8 + (threadIdx.x >> 5);
  if (e >= nui) return;
  const int lane = threadIdx.x & 31;
  const int u = iu[e], it = ii[e];
  const float wt = w[e];
  const float2 v = ((const float2*)(Ein + (size_t)it * 64))[lane];
  float* o = aggU + (size_t)u * 64 + lane * 2;
  atomicAdd(o,     v.x * wt);
  atomicAdd(o + 1, v.y * wt);
}

// ---------------------------------------------------------------------------
// Row L2-normalize; write normalized row to `cur` (if non-null) and add to res.
// One row per wave32.
// ---------------------------------------------------------------------------
__global__ __launch_bounds__(256)
void norm_res_kernel(const float* __restrict__ agg, float* __restrict__ cur,
                     float* __restrict__ res, int nrows) {
  const int row = blockIdx.x * 8 + (threadIdx.x >> 5);
  if (row >= nrows) return;
  const int lane = threadIdx.x & 31;
  const float2 v = ((const float2*)(agg + (size_t)row * 64))[lane];
  float ss = v.x * v.x + v.y * v.y;
  ss += __shfl_xor(ss, 1, 32);
  ss += __shfl_xor(ss, 2, 32);
  ss += __shfl_xor(ss, 4, 32);
  ss += __shfl_xor(ss, 8, 32);
  ss += __shfl_xor(ss, 16, 32);
  const float scale = 1.0f / fmaxf(sqrtf(ss), 1e-12f);
  float2 o; o.x = v.x * scale; o.y = v.y * scale;
  if (cur) ((float2*)(cur + (size_t)row * 64))[lane] = o;
  float* rp = res + (size_t)row * 64 + lane * 2;
  rp[0] += o.x;
  rp[1] += o.y;
}

// ---------------------------------------------------------------------------
extern "C" void kernel_launch(void* const* d_in, const int* in_sizes, int n_in,
                              void* d_out, int out_size, void* d_ws, size_t ws_size,
                              hipStream_t stream) {
  const float* user_emb     = (const float*)d_in[0];
  const float* entity_emb   = (const float*)d_in[1];
  const float* relation_emb = (const float*)d_in[2];
  const float* W_Q          = (const float*)d_in[3];
  const int*   edge_index   = (const int*)d_in[4];
  const int*   edge_type    = (const int*)d_in[5];
  const int*   inter_edge   = (const int*)d_in[6];
  const float* inter_w      = (const float*)d_in[7];

  const int n_users = in_sizes[0] / 64;
  const int n_ent   = in_sizes[1] / 64;
  const int nedge   = in_sizes[5];
  const int nui     = in_sizes[7];

  const int* head = edge_index;
  const int* tail = edge_index + nedge;
  const int* iu   = inter_edge;
  const int* ii   = inter_edge + nui;

  float* entity_res = (float*)d_out;
  float* user_res   = entity_res + (size_t)n_ent * 64;

  // ---- workspace carve (~112 MB) ----
  char* w = (char*)d_ws;
  float* P        = (float*)w;  w += (size_t)n_ent   * 64 * sizeof(float);
  float* cur      = (float*)w;  w += (size_t)n_ent   * 64 * sizeof(float);
  float* aggE     = (float*)w;  w += (size_t)n_ent   * 64 * sizeof(float);
  float* aggU     = (float*)w;  w += (size_t)n_users * 64 * sizeof(float);
  float* scorebuf = (float*)w;  w += (size_t)nedge   * 2  * sizeof(float);
  int*   maxbuf   = (int*)w;    w += (size_t)n_ent   * 2  * sizeof(int);
  float* denom    = (float*)w;  w += (size_t)n_ent   * 2  * sizeof(float);

  // residual init: res = input embeddings
  hipMemcpyAsync(entity_res, entity_emb, (size_t)n_ent   * 64 * sizeof(float),
                 hipMemcpyDeviceToDevice, stream);
  hipMemcpyAsync(user_res,   user_emb,   (size_t)n_users * 64 * sizeof(float),
                 hipMemcpyDeviceToDevice, stream);

  const float* Ein = entity_emb;
  for (int hop = 0; hop < 2; ++hop) {
    hipMemsetAsync(aggE,  0,    (size_t)n_ent   * 64 * sizeof(float), stream);
    hipMemsetAsync(aggU,  0,    (size_t)n_users * 64 * sizeof(float), stream);
    hipMemsetAsync(denom, 0,    (size_t)n_ent   * 2  * sizeof(float), stream);
    hipMemsetAsync(maxbuf, 0x80,(size_t)n_ent   * 2  * sizeof(int),   stream);
    // 0x80808080 < ordered-encoding of any representable score -> valid -inf init

    const int rowTiles = (n_ent + 15) / 16;
    proj_wmma_kernel<<<(rowTiles + 7) / 8, 256, 0, stream>>>(Ein, W_Q, P, n_ent);
    score_kernel<<<(nedge + 7) / 8, 256, 0, stream>>>(P, relation_emb, head, tail,
                                                      edge_type, scorebuf, maxbuf, nedge);
    exp_kernel<<<(nedge * 2 + 255) / 256, 256, 0, stream>>>(scorebuf, head, maxbuf,
                                                            denom, nedge);
    agg_kernel<<<(nedge + 7) / 8, 256, 0, stream>>>(Ein, relation_emb, head, tail,
                                                    edge_type, scorebuf, denom, aggE, nedge);
    user_agg_kernel<<<(nui + 7) / 8, 256, 0, stream>>>(Ein, iu, ii, inter_w, aggU, nui);
    norm_res_kernel<<<(n_ent + 7) / 8, 256, 0, stream>>>(aggE, cur, entity_res, n_ent);
    norm_res_kernel<<<(n_users + 7) / 8, 256, 0, stream>>>(aggU, nullptr, user_res, n_users);

    Ein = cur;   // next hop consumes normalized entity embeddings
  }
}